// ObjectPointwiseEncoder_33715493273851
// MI455X (gfx1250) — compile-verified
//
#include <hip/hip_runtime.h>
#include <hip/hip_bf16.h>

typedef float v2f __attribute__((ext_vector_type(2)));
typedef float v8f __attribute__((ext_vector_type(8)));

#define BIGF 1e10f

// ---------------------------------------------------------------------------
// CDNA5 async global->LDS copy (ASYNCcnt-tracked). Per-lane LDS dest address
// in a VGPR, per-lane 64-bit global source address. Inline asm per the ISA
// (portable across toolchains; bypasses the clang builtin arity differences).
// ---------------------------------------------------------------------------
__device__ __forceinline__ void async_g2l_b32(unsigned lds_byte_addr, const float* gsrc)
{
    asm volatile("global_load_async_to_lds_b32 %0, %1, off"
                 :: "v"(lds_byte_addr), "v"(gsrc)
                 : "memory");
}
__device__ __forceinline__ void wait_async0()
{
    asm volatile("s_wait_asynccnt 0x0" ::: "memory");
}

// ---------------------------------------------------------------------------
// FPS: one workgroup per batch. Sequential farthest-point sampling.
// ---------------------------------------------------------------------------
__global__ __launch_bounds__(256) void fps_kernel(const float* __restrict__ pos,
                                                  float* __restrict__ opos,
                                                  int* __restrict__ oidx,
                                                  int n, int m)
{
    const int b = blockIdx.x;
    const float* P = pos + (size_t)b * n * 3;
    float* OP = opos + (size_t)b * m * 3;
    int* OI = oidx + (size_t)b * m;
    const int tid = threadIdx.x;

    float mind[16];
    const int cnt = (n + 255) >> 8;
    for (int i = 0; i < 16; ++i) mind[i] = BIGF;

    __shared__ float sv[256];
    __shared__ int   si[256];
    __shared__ float cp[3];

    int last = 0;
    if (tid == 0) OI[0] = 0;
    if (tid < 3) OP[tid] = P[tid];

    for (int s = 1; s < m; ++s) {
        if (tid < 3) cp[tid] = P[(size_t)last * 3 + tid];
        __syncthreads();
        float px = cp[0], py = cp[1], pz = cp[2];
        float best = -1.0f; int bi = 0;
        for (int i = 0; i < cnt; ++i) {
            int j = tid + (i << 8);
            if (j < n) {
                float dx = P[j*3+0]-px, dy = P[j*3+1]-py, dz = P[j*3+2]-pz;
                float d = dx*dx + dy*dy + dz*dz;
                if (d < mind[i]) mind[i] = d;
                if (mind[i] > best) { best = mind[i]; bi = j; }
            }
        }
        sv[tid] = best; si[tid] = bi;
        __syncthreads();
        for (int off = 128; off > 0; off >>= 1) {
            if (tid < off) {
                float ov = sv[tid+off]; int oi = si[tid+off];
                if (ov > sv[tid] || (ov == sv[tid] && oi < si[tid])) { sv[tid] = ov; si[tid] = oi; }
            }
            __syncthreads();
        }
        last = si[0];
        if (tid == 0) OI[s] = last;
        if (tid < 3) OP[(size_t)s*3 + tid] = P[(size_t)last*3 + tid];
        __syncthreads();
    }
}

// ---------------------------------------------------------------------------
// Radius grouping: <=64 nearest within r. One block per (center, batch).
// ---------------------------------------------------------------------------
__global__ __launch_bounds__(256) void radius_topk_kernel(const float* __restrict__ spos,
                                                          const float* __restrict__ cpos,
                                                          int* __restrict__ nidx,
                                                          int n, int m, float r2)
{
    const int c = blockIdx.x, b = blockIdx.y;
    const float* P  = spos + (size_t)b * n * 3;
    const float* CP = cpos + ((size_t)b * m + c) * 3;
    int* NI = nidx + ((size_t)b * m + c) * 64;
    const int tid = threadIdx.x;

    __shared__ float d2s[4096];
    __shared__ float sv[256];
    __shared__ int   si[256];
    __shared__ int   stop;

    float cx = CP[0], cy = CP[1], cz = CP[2];
    for (int j = tid; j < n; j += 256) {
        float dx = P[j*3]-cx, dy = P[j*3+1]-cy, dz = P[j*3+2]-cz;
        float d = dx*dx + dy*dy + dz*dz;
        d2s[j] = (d <= r2) ? d : BIGF;
    }
    __syncthreads();

    for (int k = 0; k < 64; ++k) {
        float best = BIGF; int bi = n;
        for (int j = tid; j < n; j += 256) {
            float v = d2s[j];
            if (v < best) { best = v; bi = j; }
        }
        sv[tid] = best; si[tid] = bi;
        __syncthreads();
        for (int off = 128; off > 0; off >>= 1) {
            if (tid < off) {
                float ov = sv[tid+off]; int oi = si[tid+off];
                if (ov < sv[tid] || (ov == sv[tid] && oi < si[tid])) { sv[tid] = ov; si[tid] = oi; }
            }
            __syncthreads();
        }
        if (tid == 0) {
            if (sv[0] < BIGF * 0.5f) { NI[k] = si[0]; d2s[si[0]] = BIGF; stop = 0; }
            else { for (int kk = k; kk < 64; ++kk) NI[kk] = -1; stop = 1; }
        }
        __syncthreads();
        if (stop) break;
    }
}

// ---------------------------------------------------------------------------
// fp32 WMMA 16x16 tile helper: K multiple of 4, fully unguarded inner loop.
// A layout: lane m=L&15, lane-half picks K pair; B: lane n=L&15, same pattern.
// ---------------------------------------------------------------------------
__device__ __forceinline__ v8f wmma_f32_tile_main(const float* __restrict__ A, int lda,
                                                  const float* __restrict__ W, int ldw,
                                                  int Kmain, int mt, int nt, int lane)
{
    const int half = lane >> 4;
    const int ml = mt * 16 + (lane & 15);
    const int nc = nt * 16 + (lane & 15);
    const float* Arow = A + (size_t)ml * lda;
    v8f acc = {0.f,0.f,0.f,0.f,0.f,0.f,0.f,0.f};
    for (int k0 = 0; k0 < Kmain; k0 += 4) {
        int ka = k0 + 2 * half;
        v2f av, bv;
        av.x = Arow[ka];
        av.y = Arow[ka + 1];
        bv.x = W[(size_t)ka * ldw + nc];
        bv.y = W[((size_t)ka + 1) * ldw + nc];
        acc = __builtin_amdgcn_wmma_f32_16x16x4_f32(false, av, false, bv,
                                                    (short)0, acc, false, false);
    }
    return acc;
}

// ---------------------------------------------------------------------------
// Fused SA module: gather neighbors -> 3-layer MLP (WMMA) -> masked row max.
// One block (8 waves) per (center, batch). Activations stay in LDS. The
// neighbor x-feature gather and the zero-padded W0-tail staging use the
// CDNA5 async global->LDS path; pos-deltas / zero-fill use disjoint ds writes.
// ---------------------------------------------------------------------------
template<int CXF, int C1, int C2, int C3>
__global__ __launch_bounds__(256) void sa_conv_kernel(
    const float* __restrict__ xf,   const float* __restrict__ spos,
    const float* __restrict__ cpos, const int* __restrict__ nidx,
    const float* __restrict__ W0, const float* __restrict__ B0,
    const float* __restrict__ W1, const float* __restrict__ B1,
    const float* __restrict__ W2, const float* __restrict__ B2,
    float* __restrict__ out, int n, int m)
{
    constexpr int CIN  = CXF + 3;
    constexpr int CINP = (CIN + 3) & ~3;
    constexpr int KM1  = CIN & ~3;          // main K, multiple of 4
    constexpr int LDA  = (CINP > C2) ? CINP : C2;

    const int c = blockIdx.x, b = blockIdx.y;
    xf   += (size_t)b * n * CXF;
    spos += (size_t)b * n * 3;
    const float* CP = cpos + ((size_t)b * m + c) * 3;
    const int*   NI = nidx + ((size_t)b * m + c) * 64;
    float*       O  = out  + ((size_t)b * m + c) * C3;

    __shared__ float Abuf[64 * LDA];   // F (ld=CINP), later reused as H2 (ld=C2)
    __shared__ float H1[64 * C1];
    __shared__ float Wt[4 * C1];       // zero-padded tail rows of W0
    __shared__ int   nb[64];
    __shared__ float cx[3];

    const int tid  = threadIdx.x;
    const int wave = tid >> 5;
    const int lane = tid & 31;

    if (tid < 64) nb[tid] = NI[tid];
    if (tid < 3)  cx[tid] = CP[tid];
    __syncthreads();

    // gather features: [x[nidx] | pos[nidx]-center | zero pad]
    // x-features: async global->LDS copy; the rest: ds writes (disjoint dwords)
    for (int i = tid; i < 64 * CINP; i += 256) {
        int row = i / CINP, col = i - row * CINP;
        int id = nb[row];
        if (id >= 0 && col < CXF) {
            async_g2l_b32((unsigned)(size_t)&Abuf[i], xf + (size_t)id * CXF + col);
        } else {
            float v = 0.f;
            if (id >= 0 && col < CIN) v = spos[(size_t)id * 3 + (col - CXF)] - cx[col - CXF];
            Abuf[i] = v;
        }
    }
    // stage zero-padded W0 tail rows (KM1..KM1+3) into LDS
    if (CIN & 3) {
        for (int i = tid; i < 4 * C1; i += 256) {
            int kt = i / C1, col = i - kt * C1;
            int krow = KM1 + kt;
            if (krow < CIN) async_g2l_b32((unsigned)(size_t)&Wt[i], W0 + (size_t)krow * C1 + col);
            else            Wt[i] = 0.f;
        }
    }
    wait_async0();
    __syncthreads();

    // layer 1: H1 = relu(F @ W0 + b0); tail B-fragment comes from LDS (no guards)
    constexpr int NT1 = C1 / 16;
    for (int t = wave; t < 4 * NT1; t += 8) {
        int mt = t / NT1, nt = t % NT1;
        int half = lane >> 4, ml = mt * 16 + (lane & 15), col = nt * 16 + (lane & 15);
        v8f acc = wmma_f32_tile_main(Abuf, CINP, W0, C1, KM1, mt, nt, lane);
        if constexpr ((CIN & 3) != 0) {
            int ka = KM1 + 2 * half;
            v2f av, bv;
            av.x = Abuf[ml * CINP + ka];       // zero pad cols
            av.y = Abuf[ml * CINP + ka + 1];
            bv.x = Wt[(2 * half)     * C1 + col];
            bv.y = Wt[(2 * half + 1) * C1 + col];
            acc = __builtin_amdgcn_wmma_f32_16x16x4_f32(false, av, false, bv,
                                                        (short)0, acc, false, false);
        }
        float bv0 = B0[col];
        for (int r = 0; r < 8; ++r)
            H1[(mt*16 + r + 8*half) * C1 + col] = fmaxf(acc[r] + bv0, 0.f);
    }
    __syncthreads();

    // layer 2: H2 (in Abuf) = relu(H1 @ W1 + b1)
    constexpr int NT2 = C2 / 16;
    for (int t = wave; t < 4 * NT2; t += 8) {
        int mt = t / NT2, nt = t % NT2;
        v8f acc = wmma_f32_tile_main(H1, C1, W1, C2, C1, mt, nt, lane);
        int half = lane >> 4, col = nt * 16 + (lane & 15);
        float bv = B1[col];
        for (int r = 0; r < 8; ++r)
            Abuf[(mt*16 + r + 8*half) * C2 + col] = fmaxf(acc[r] + bv, 0.f);
    }
    __syncthreads();

    // layer 3 fused with masked max over the 64 neighbor rows
    constexpr int NT3 = C3 / 16;
    for (int nt = wave; nt < NT3; nt += 8) {
        int half = lane >> 4, col = nt * 16 + (lane & 15);
        float bv = B2[col];
        v8f cmax = {-BIGF,-BIGF,-BIGF,-BIGF,-BIGF,-BIGF,-BIGF,-BIGF};
        for (int mt = 0; mt < 4; ++mt) {
            v8f acc = wmma_f32_tile_main(Abuf, C2, W2, C3, C2, mt, nt, lane);
            for (int r = 0; r < 8; ++r) {
                int row = mt*16 + r + 8*half;
                float v = (nb[row] >= 0) ? (acc[r] + bv) : -BIGF;
                cmax[r] = fmaxf(cmax[r], v);
            }
        }
        float mv = cmax[0];
        for (int r = 1; r < 8; ++r) mv = fmaxf(mv, cmax[r]);
        mv = fmaxf(mv, __shfl_xor(mv, 16, 32));
        if (lane < 16) O[col] = mv;
    }
}

// ---------------------------------------------------------------------------
// Generic fp32 WMMA GEMM: C = act(A @ W + bias). One wave per block; each
// wave computes NTW adjacent 16x16 N-tiles so one A fragment feeds NTW WMMAs.
// ---------------------------------------------------------------------------
template<int NTW>
__global__ __launch_bounds__(32) void wmma_gemm_kernel(
    const float* __restrict__ A, int lda, size_t strideA,
    const float* __restrict__ W, const float* __restrict__ bias,
    float* __restrict__ C, int ldc, size_t strideC,
    int M, int K, int N, int relu)
{
    const int mt = blockIdx.x, ntg = blockIdx.y, b = blockIdx.z;
    A += (size_t)b * strideA;
    C += (size_t)b * strideC;
    const int lane = threadIdx.x;
    const int half = lane >> 4;
    int ml = mt * 16 + (lane & 15);
    if (ml >= M) ml = M - 1;
    const int nc0 = ntg * (16 * NTW) + (lane & 15);
    const float* Arow = A + (size_t)ml * lda;

    v8f acc[NTW];
#pragma unroll
    for (int t = 0; t < NTW; ++t) acc[t] = (v8f){0.f,0.f,0.f,0.f,0.f,0.f,0.f,0.f};

    const int Kmain = K & ~3;
    for (int k0 = 0; k0 < Kmain; k0 += 4) {
        int ka = k0 + 2 * half;
        v2f av;
        av.x = Arow[ka];
        av.y = Arow[ka + 1];
        const float* Wr0 = W + (size_t)ka * N + nc0;
        const float* Wr1 = Wr0 + N;
#pragma unroll
        for (int t = 0; t < NTW; ++t) {
            v2f bv;
            bv.x = Wr0[t * 16];
            bv.y = Wr1[t * 16];
            acc[t] = __builtin_amdgcn_wmma_f32_16x16x4_f32(false, av, false, bv,
                                                           (short)0, acc[t], false, false);
        }
    }
    if (K & 3) {
        int ka = Kmain + 2 * half;
        v2f av;
        av.x = Arow[ka];          // zero pad cols in A
        av.y = Arow[ka + 1];
        bool gx = (ka < K), gy = (ka + 1 < K);
#pragma unroll
        for (int t = 0; t < NTW; ++t) {
            v2f bv;
            bv.x = gx ? W[(size_t)ka * N + nc0 + t * 16]       : 0.f;
            bv.y = gy ? W[((size_t)ka + 1) * N + nc0 + t * 16] : 0.f;
            acc[t] = __builtin_amdgcn_wmma_f32_16x16x4_f32(false, av, false, bv,
                                                           (short)0, acc[t], false, false);
        }
    }

#pragma unroll
    for (int t = 0; t < NTW; ++t) {
        int nc = nc0 + t * 16;
        float bvv = bias[nc];
        for (int r = 0; r < 8; ++r) {
            int mm = mt * 16 + r + 8 * half;
            if (mm < M) {
                float v = acc[t][r] + bvv;
                if (relu) v = fmaxf(v, 0.f);
                C[(size_t)mm * ldc + nc] = v;
            }
        }
    }
}

// ---------------------------------------------------------------------------
// Concat two row-features (A may be row-broadcast via rsA=0); zero pad to ldd.
// ---------------------------------------------------------------------------
__global__ __launch_bounds__(256) void concat2_kernel(
    const float* __restrict__ A, int ca, int rsA, size_t sA,
    const float* __restrict__ Bx, int cb, size_t sB,
    float* __restrict__ D, int ldd, size_t sD, int rows)
{
    const int b = blockIdx.y;
    size_t i = (size_t)blockIdx.x * 256 + threadIdx.x;
    size_t tot = (size_t)rows * ldd;
    if (i >= tot) return;
    int r = (int)(i / ldd), col = (int)(i % ldd);
    float v = 0.f;
    if (col < ca)            v = A[sA * b + (size_t)r * rsA + col];
    else if (col < ca + cb)  v = Bx[sB * b + (size_t)r * cb + (col - ca)];
    D[sD * b + i] = v;
}

// ---------------------------------------------------------------------------
// kNN(k=3) inverse-d2-weighted interpolation fused with concat + zero pad.
// One thread per target point.
// ---------------------------------------------------------------------------
__global__ __launch_bounds__(256) void knn3_concat_kernel(
    const float* __restrict__ srcF, const float* __restrict__ srcP,
    const float* __restrict__ tgtP, const float* __restrict__ extraF,
    float* __restrict__ D, int ns, int nt, int C, int CE, int ldd)
{
    const int b = blockIdx.y;
    int t = blockIdx.x * 256 + threadIdx.x;
    if (t >= nt) return;
    srcF   += (size_t)b * ns * C;
    srcP   += (size_t)b * ns * 3;
    tgtP   += (size_t)b * nt * 3;
    extraF += (size_t)b * nt * CE;
    D      += (size_t)b * nt * ldd + (size_t)t * ldd;

    float tx = tgtP[t*3], ty = tgtP[t*3+1], tz = tgtP[t*3+2];
    float bd0 = BIGF, bd1 = BIGF, bd2 = BIGF;
    int   bi0 = 0,    bi1 = 0,    bi2 = 0;
    for (int j = 0; j < ns; ++j) {
        float dx = srcP[j*3]-tx, dy = srcP[j*3+1]-ty, dz = srcP[j*3+2]-tz;
        float d = dx*dx + dy*dy + dz*dz;
        if (d < bd0)      { bd2=bd1; bi2=bi1; bd1=bd0; bi1=bi0; bd0=d; bi0=j; }
        else if (d < bd1) { bd2=bd1; bi2=bi1; bd1=d; bi1=j; }
        else if (d < bd2) { bd2=d; bi2=j; }
    }
    float w0 = 1.f / fmaxf(bd0, 1e-16f);
    float w1 = 1.f / fmaxf(bd1, 1e-16f);
    float w2 = 1.f / fmaxf(bd2, 1e-16f);
    float wsum = w0 + w1 + w2;
    const float* f0 = srcF + (size_t)bi0 * C;
    const float* f1 = srcF + (size_t)bi1 * C;
    const float* f2 = srcF + (size_t)bi2 * C;
    for (int cc = 0; cc < C; ++cc)
        D[cc] = (w0*f0[cc] + w1*f1[cc] + w2*f2[cc]) / wsum;
    for (int cc = 0; cc < CE; ++cc)
        D[C + cc] = extraF[(size_t)t * CE + cc];
    for (int cc = C + CE; cc < ldd; ++cc) D[cc] = 0.f;
}

// ---------------------------------------------------------------------------
// Column-wise max over rows (global max pool).
// ---------------------------------------------------------------------------
__global__ __launch_bounds__(256) void colmax_kernel(const float* __restrict__ A,
                                                     float* __restrict__ O,
                                                     int rows, int cols)
{
    const int b = blockIdx.y;
    int cc = blockIdx.x * 256 + threadIdx.x;
    if (cc >= cols) return;
    A += (size_t)b * rows * cols;
    float mv = -BIGF;
    for (int r = 0; r < rows; ++r) mv = fmaxf(mv, A[(size_t)r * cols + cc]);
    O[(size_t)b * cols + cc] = mv;
}

// ---------------------------------------------------------------------------
extern "C" void kernel_launch(void* const* d_in, const int* in_sizes, int n_in,
                              void* d_out, int out_size, void* d_ws, size_t ws_size,
                              hipStream_t stream)
{
    (void)in_sizes; (void)n_in; (void)out_size; (void)ws_size;
    const int Bn = 16, Np = 4096, Na = 820, Nb = 205, Kn = 64;

    const float* x0 = (const float*)d_in[0];
    const float* p0 = (const float*)d_in[1];
    const float* W[16]; const float* Bi[16];
    for (int i = 0; i < 16; ++i) { W[i] = (const float*)d_in[2 + 2*i]; Bi[i] = (const float*)d_in[3 + 2*i]; }

    char* ws = (char*)d_ws;
    size_t o = 0;
    auto take = [&](size_t elems) -> char* {
        char* p = ws + o;
        o += (elems * sizeof(float) + 255) & ~(size_t)255;
        return p;
    };
    float* pos1 = (float*)take((size_t)Bn * Na * 3);
    float* pos2 = (float*)take((size_t)Bn * Nb * 3);
    int*   idx1 = (int*)  take((size_t)Bn * Na);
    int*   idx2 = (int*)  take((size_t)Bn * Nb);
    int*   ni1  = (int*)  take((size_t)Bn * Na * Kn);
    int*   ni2  = (int*)  take((size_t)Bn * Nb * Kn);
    float* x1   = (float*)take((size_t)Bn * Na * 128);
    float* x2   = (float*)take((size_t)Bn * Nb * 256);
    float* h3   = (float*)take((size_t)Bn * Nb * 256);
    float* h2   = (float*)take((size_t)Bn * Na * 128);
    float* xg   = (float*)take((size_t)Bn * 1024);
    char* pool = ws + o;   // transient region, reused stage-to-stage (stream-ordered)

    // stage: global SA temporaries
    float* g0 = (float*)pool;
    float* g1 = g0 + (size_t)Bn * Nb * 260;
    float* g2 = g1 + (size_t)Bn * Nb * 256;
    float* g3 = g2 + (size_t)Bn * Nb * 512;
    // stage: FP3
    float* a3 = (float*)pool;
    float* t3 = a3 + (size_t)Bn * Nb * 1280;
    // stage: FP2
    float* a2 = (float*)pool;
    float* t2 = a2 + (size_t)Bn * Na * 384;
    // stage: FP1
    float* a1  = (float*)pool;
    float* t1a = a1  + (size_t)Bn * Np * 132;
    float* t1b = t1a + (size_t)Bn * Np * 128;
    float* outp = (float*)d_out;

    // --- SA module 1: 4096 -> 820, r=0.2, MLP 6->64->64->128 ---
    fps_kernel<<<Bn, 256, 0, stream>>>(p0, pos1, idx1, Np, Na);
    radius_topk_kernel<<<dim3(Na, Bn), 256, 0, stream>>>(p0, pos1, ni1, Np, Na, 0.04f);
    sa_conv_kernel<3, 64, 64, 128><<<dim3(Na, Bn), 256, 0, stream>>>(
        x0, p0, pos1, ni1, W[0], Bi[0], W[1], Bi[1], W[2], Bi[2], x1, Np, Na);

    // --- SA module 2: 820 -> 205, r=0.4, MLP 131->128->128->256 ---
    fps_kernel<<<Bn, 256, 0, stream>>>(pos1, pos2, idx2, Na, Nb);
    radius_topk_kernel<<<dim3(Nb, Bn), 256, 0, stream>>>(pos1, pos2, ni2, Na, Nb, 0.16f);
    sa_conv_kernel<128, 128, 128, 256><<<dim3(Nb, Bn), 256, 0, stream>>>(
        x1, pos1, pos2, ni2, W[3], Bi[3], W[4], Bi[4], W[5], Bi[5], x2, Na, Nb);

    // --- Global SA: MLP 259->256->512->1024 then max over 205 rows ---
    {
        size_t tot = (size_t)Nb * 260;
        concat2_kernel<<<dim3((unsigned)((tot + 255) / 256), Bn), 256, 0, stream>>>(
            x2, 256, 256, (size_t)Nb * 256, pos2, 3, (size_t)Nb * 3,
            g0, 260, (size_t)Nb * 260, Nb);
    }
    wmma_gemm_kernel<4><<<dim3(13, 4,  Bn), 32, 0, stream>>>(g0, 260, (size_t)Nb*260, W[6], Bi[6], g1, 256,  (size_t)Nb*256,  Nb, 259, 256, 1);
    wmma_gemm_kernel<4><<<dim3(13, 8,  Bn), 32, 0, stream>>>(g1, 256, (size_t)Nb*256, W[7], Bi[7], g2, 512,  (size_t)Nb*512,  Nb, 256, 512, 1);
    wmma_gemm_kernel<4><<<dim3(13, 16, Bn), 32, 0, stream>>>(g2, 512, (size_t)Nb*512, W[8], Bi[8], g3, 1024, (size_t)Nb*1024, Nb, 512, 1024, 0);
    colmax_kernel<<<dim3(4, Bn), 256, 0, stream>>>(g3, xg, Nb, 1024);

    // --- FP3: [broadcast(xg) | x2] -> MLP 1280->256->256 ---
    {
        size_t tot = (size_t)Nb * 1280;
        concat2_kernel<<<dim3((unsigned)((tot + 255) / 256), Bn), 256, 0, stream>>>(
            xg, 1024, 0, (size_t)1024, x2, 256, (size_t)Nb * 256,
            a3, 1280, (size_t)Nb * 1280, Nb);
    }
    wmma_gemm_kernel<4><<<dim3(13, 4, Bn), 32, 0, stream>>>(a3, 1280, (size_t)Nb*1280, W[9],  Bi[9],  t3, 256, (size_t)Nb*256, Nb, 1280, 256, 1);
    wmma_gemm_kernel<4><<<dim3(13, 4, Bn), 32, 0, stream>>>(t3, 256,  (size_t)Nb*256,  W[10], Bi[10], h3, 256, (size_t)Nb*256, Nb, 256,  256, 0);

    // --- FP2: knn3(h3: pos2 -> pos1) | x1 -> MLP 384->256->128 ---
    knn3_concat_kernel<<<dim3((Na + 255) / 256, Bn), 256, 0, stream>>>(
        h3, pos2, pos1, x1, a2, Nb, Na, 256, 128, 384);
    wmma_gemm_kernel<4><<<dim3((Na + 15) / 16, 4, Bn), 32, 0, stream>>>(a2, 384, (size_t)Na*384, W[11], Bi[11], t2, 256, (size_t)Na*256, Na, 384, 256, 1);
    wmma_gemm_kernel<4><<<dim3((Na + 15) / 16, 2, Bn), 32, 0, stream>>>(t2, 256, (size_t)Na*256, W[12], Bi[12], h2, 128, (size_t)Na*128, Na, 256, 128, 0);

    // --- FP1: knn3(h2: pos1 -> pos0) | x0 -> MLP 131->128->128->128 ---
    knn3_concat_kernel<<<dim3((Np + 255) / 256, Bn), 256, 0, stream>>>(
        h2, pos1, p0, x0, a1, Na, Np, 128, 3, 132);
    wmma_gemm_kernel<4><<<dim3(Np / 16, 2, Bn), 32, 0, stream>>>(a1,  132, (size_t)Np*132, W[13], Bi[13], t1a,  128, (size_t)Np*128, Np, 131, 128, 1);
    wmma_gemm_kernel<4><<<dim3(Np / 16, 2, Bn), 32, 0, stream>>>(t1a, 128, (size_t)Np*128, W[14], Bi[14], t1b,  128, (size_t)Np*128, Np, 128, 128, 1);
    wmma_gemm_kernel<4><<<dim3(Np / 16, 2, Bn), 32, 0, stream>>>(t1b, 128, (size_t)Np*128, W[15], Bi[15], outp, 128, (size_t)Np*128, Np, 128, 128, 0);
}